// GraphEncoder_19000935317646
// MI455X (gfx1250) — compile-verified
//
#include <hip/hip_runtime.h>
#include <math.h>

typedef __attribute__((ext_vector_type(2))) float v2f;
typedef __attribute__((ext_vector_type(8))) float v8f;

#define DIM 96
#define PITCH 98   // LDS row pitch: 98 mod 64 = 34 -> lanes 0..15 map to distinct banks, keeps 8B align

// ---------------------------------------------------------------------------
// degree: deg[dst[e]] += 1
__global__ void k_degree(const int* __restrict__ dst, int* __restrict__ deg, int n_edges) {
    int e = blockIdx.x * blockDim.x + threadIdx.x;
    if (e < n_edges) atomicAdd(&deg[dst[e]], 1);
}

// norm[n] = max(deg,1)^(-1/2)
__global__ void k_norm(const int* __restrict__ deg, float* __restrict__ nrm, int n_nodes) {
    int n = blockIdx.x * blockDim.x + threadIdx.x;
    if (n < n_nodes) nrm[n] = rsqrtf(fmaxf((float)deg[n], 1.0f));
}

// ---------------------------------------------------------------------------
// out[n][j] = (sum_k in[n][k] * W[j][k] + b[j]) * norm[n]
// One block = 16 rows x 96 cols. 6 waves; wave w computes the 16x16 tile at
// columns [16w, 16w+16) using chained V_WMMA_F32_16X16X4_F32 over K=96.
// Safe to run in place (in == out): block reads only its own rows into LDS
// (barrier) before writing only its own rows.
__global__ __launch_bounds__(192) void k_gemm(const float* __restrict__ in,
                                              const float* __restrict__ W,
                                              const float* __restrict__ bias,
                                              const float* __restrict__ nrm,
                                              float* __restrict__ out,
                                              int n_nodes) {
    __shared__ float sW[DIM * PITCH];   // W[j][k] at sW[j*PITCH + k]
    __shared__ float sA[16 * PITCH];    // A rows
    __shared__ float sB[DIM];
    __shared__ float sN[16];

    const int tid = threadIdx.x;
    const int row_base = blockIdx.x * 16;

    for (int i = tid; i < DIM * DIM; i += 192) {
        int r = i / DIM, c = i - r * DIM;
        sW[r * PITCH + c] = W[i];
    }
    for (int i = tid; i < 16 * DIM; i += 192) {
        int r = i / DIM, c = i - r * DIM;
        int gr = row_base + r;
        sA[r * PITCH + c] = (gr < n_nodes) ? in[(size_t)gr * DIM + c] : 0.0f;
    }
    if (tid < DIM) sB[tid] = bias[tid];
    if (tid < 16) {
        int gr = row_base + tid;
        sN[tid] = (gr < n_nodes) ? nrm[gr] : 0.0f;
    }
    __syncthreads();

    const int lane  = tid & 31;
    const int wave  = tid >> 5;        // 0..5
    const int cb    = wave * 16;       // column tile base
    const int lo    = lane & 15;       // M (for A) / N (for B,C)
    const int kgrp  = (lane >> 4) * 2; // K sub-offset: 0 for lanes 0-15, 2 for 16-31

    v8f c = {};
    #pragma unroll
    for (int kk = 0; kk < DIM; kk += 4) {
        v2f a, b;
        a.x = sA[lo * PITCH + kk + kgrp];
        a.y = sA[lo * PITCH + kk + kgrp + 1];
        b.x = sW[(cb + lo) * PITCH + kk + kgrp];   // B[k][n] = W^T[k][n] = W[n][k]
        b.y = sW[(cb + lo) * PITCH + kk + kgrp + 1];
        c = __builtin_amdgcn_wmma_f32_16x16x4_f32(false, a, false, b,
                                                  (short)0, c, false, false);
    }

    const int col = cb + lo;
    #pragma unroll
    for (int r = 0; r < 8; ++r) {
        int row_l = r + ((lane >> 4) << 3);   // C VGPR r: M=r (lanes 0-15) / M=r+8 (lanes 16-31)
        int row = row_base + row_l;
        if (row < n_nodes)
            out[(size_t)row * DIM + col] = (c[r] + sB[col]) * sN[row_l];
    }
}

// ---------------------------------------------------------------------------
// acc[dst[e]][f] += feat[src[e]][f] * factor[e]   (one thread per (edge,feature))
__global__ void k_scatter(const float* __restrict__ feat,
                          const float* __restrict__ factor,
                          const int* __restrict__ src,
                          const int* __restrict__ dst,
                          float* __restrict__ acc,
                          int n_edges) {
    int idx = blockIdx.x * blockDim.x + threadIdx.x;
    int e = idx / DIM;
    if (e >= n_edges) return;
    int f = idx - e * DIM;
    float v = feat[(size_t)src[e] * DIM + f] * factor[e];
    atomicAdd(&acc[(size_t)dst[e] * DIM + f], v);
}

// h[i] = tanh(acc[i]); acc[i] = 0  (prepare for next propagation round)
__global__ void k_tanh_reset(float* __restrict__ h,
                             float* __restrict__ acc, int count) {
    int i = blockIdx.x * blockDim.x + threadIdx.x;
    if (i < count) {
        h[i] = tanhf(acc[i]);
        acc[i] = 0.0f;
    }
}

// per-feature partial sums of tanh(acc) over nodes
__global__ void k_pool(const float* __restrict__ acc,
                       float* __restrict__ psum, int n_nodes) {
    int f = threadIdx.x;               // 96 threads
    float s = 0.0f;
    for (int n = blockIdx.x; n < n_nodes; n += gridDim.x)
        s += tanhf(acc[(size_t)n * DIM + f]);
    atomicAdd(&psum[f], s);
}

__global__ void k_final(const float* __restrict__ psum,
                        float* __restrict__ out, int n_nodes) {
    int f = threadIdx.x;
    if (f < DIM) out[f] = tanhf(psum[f] / (float)n_nodes);
}

// ---------------------------------------------------------------------------
extern "C" void kernel_launch(void* const* d_in, const int* in_sizes, int n_in,
                              void* d_out, int out_size, void* d_ws, size_t ws_size,
                              hipStream_t stream) {
    const float* inputs = (const float*)d_in[0];
    const float* W1     = (const float*)d_in[1];
    const float* b1     = (const float*)d_in[2];
    const float* W2     = (const float*)d_in[3];
    const float* b2     = (const float*)d_in[4];
    const float* factor = (const float*)d_in[5];
    const int*   src    = (const int*)d_in[6];
    const int*   dst    = (const int*)d_in[7];
    float* out = (float*)d_out;

    const int n_nodes = in_sizes[0] / DIM;
    const int n_edges = in_sizes[5];

    // workspace layout
    float* F    = (float*)d_ws;                       // N x 96 features
    float* AC   = F + (size_t)n_nodes * DIM;          // N x 96 accumulator
    float* nrm  = AC + (size_t)n_nodes * DIM;         // N
    int*   deg  = (int*)(nrm + n_nodes);              // N
    float* psum = (float*)(deg + n_nodes);            // 96

    hipMemsetAsync(AC,   0, (size_t)n_nodes * DIM * sizeof(float), stream);
    hipMemsetAsync(deg,  0, (size_t)n_nodes * sizeof(int), stream);
    hipMemsetAsync(psum, 0, DIM * sizeof(float), stream);

    const int ew_total = n_edges * DIM;               // 76.8M, fits int
    dim3 eblk(256), egrd((ew_total + 255) / 256);
    dim3 gblk(192), ggrd((n_nodes + 15) / 16);

    k_degree<<<(n_edges + 255) / 256, 256, 0, stream>>>(dst, deg, n_edges);
    k_norm  <<<(n_nodes + 255) / 256, 256, 0, stream>>>(deg, nrm, n_nodes);

    // layer 1
    k_gemm<<<ggrd, gblk, 0, stream>>>(inputs, W1, b1, nrm, F, n_nodes);
    k_scatter<<<egrd, eblk, 0, stream>>>(F, factor, src, dst, AC, n_edges);
    k_tanh_reset<<<((n_nodes * DIM) + 255) / 256, 256, 0, stream>>>(F, AC, n_nodes * DIM);

    // layer 2 (in-place GEMM on F is safe: block-local rows, barrier before store)
    k_gemm<<<ggrd, gblk, 0, stream>>>(F, W2, b2, nrm, F, n_nodes);
    k_scatter<<<egrd, eblk, 0, stream>>>(F, factor, src, dst, AC, n_edges);

    // pooled mean -> tanh
    k_pool <<<256, DIM, 0, stream>>>(AC, psum, n_nodes);
    k_final<<<1, DIM, 0, stream>>>(psum, out, n_nodes);
}